// GCN_20822001451081
// MI455X (gfx1250) — compile-verified
//
#include <hip/hip_runtime.h>
#include <hip/hip_bf16.h>
#include <math.h>
#include <stdint.h>

typedef __attribute__((ext_vector_type(2))) float v2f;
typedef __attribute__((ext_vector_type(4))) float v4f;
typedef __attribute__((ext_vector_type(8))) float v8f;
typedef __attribute__((ext_vector_type(4))) unsigned int u32x4;
typedef __attribute__((ext_vector_type(4))) int i32x4;
typedef __attribute__((ext_vector_type(8))) int i32x8;

#define FIN  512
#define HID  16
#define NCLS 7
#define TPAD 8     // padded row stride for the 7-wide layer-2 features
#define KCHUNK 64                 // K columns staged per TDM transfer
#define NCHUNK (FIN / KCHUNK)     // 8
#define RSTRIDE 66                // LDS floats per row: 64 data + 2 pad (bank-conflict-free b64)
#define G1_WAVES 4
#define G1_BLOCK (G1_WAVES * 32)

#if defined(__has_builtin)
#if __has_builtin(__builtin_amdgcn_tensor_load_to_lds)
#define USE_TDM 1
#endif
#endif
#ifndef USE_TDM
#define USE_TDM 0
#endif

// ---------------- degree / normalization ----------------

__global__ void k_init_deg(float* __restrict__ deg, int n) {
    int i = blockIdx.x * blockDim.x + threadIdx.x;
    if (i < n) deg[i] = 1.0f;   // self-loop weight
}

__global__ void k_deg_edges(const long long* __restrict__ ei,
                            const float* __restrict__ w,
                            float* __restrict__ deg, int E) {
    int e = blockIdx.x * blockDim.x + threadIdx.x;
    if (e < E) {
        int d = (int)ei[(size_t)E + e];
        atomicAdd(&deg[d], w[e]);
    }
}

__global__ void k_dinv(float* __restrict__ deg, int n) {
    int i = blockIdx.x * blockDim.x + threadIdx.x;
    if (i < n) {
        float d = deg[i];
        deg[i] = (d > 0.0f) ? rsqrtf(d) : 0.0f;
    }
}

// ---------------- TDM tile loader: 16 rows x KCHUNK f32 cols of x -> LDS ----------------
// D# per CDNA5 ISA:
//  group0: count=1 | lds_addr | global_addr[56:0] | type=2
//  group1: data_size=4B, pad_enable (2 DWORDs every 64 DWORDs -> LDS row stride 66 f32),
//          tensor_dim0=512, tensor_dim1=n, tile_dim0=64, tile_dim1=rows, dim0_stride=512
#if USE_TDM
__device__ __forceinline__ void tdm_load_tile16(const float* gsrc, unsigned lds_byte_addr,
                                                int n_total, int rows) {
    unsigned long long ga = (unsigned long long)(uintptr_t)gsrc;
    unsigned g00 = 1u;                                                  // count=1
    unsigned g01 = lds_byte_addr;                                       // LDS dest (bytes)
    unsigned g02 = (unsigned)ga;                                        // global addr lo
    unsigned g03 = (unsigned)((ga >> 32) & 0x01FFFFFFu) | (2u << 30);   // addr hi | type=2
    unsigned g10 = (2u << 16)      // data_size = 4B
                 | (1u << 20)      // pad_enable
                 | (5u << 22)      // pad_interval: every 64 DWORDs (256B = one row)
                 | (1u << 25);     // pad_amount: 2 DWORDs
    unsigned g11 = ((unsigned)FIN) << 16;                               // tensor_dim0 lo16
    unsigned g12 = ((unsigned)n_total & 0xFFFFu) << 16;                 // dim0 hi16=0 | tensor_dim1 lo16
    unsigned g13 = ((unsigned)n_total >> 16) | (((unsigned)KCHUNK) << 16); // dim1 hi16 | tile_dim0
    unsigned g14 = (unsigned)rows;                                      // tile_dim1 (tile_dim2=0)
    unsigned g15 = (unsigned)FIN;                                       // tensor_dim0_stride lo32
    unsigned g16 = 0u, g17 = 0u;

    u32x4 g0 = { (unsigned)__builtin_amdgcn_readfirstlane((int)g00),
                 (unsigned)__builtin_amdgcn_readfirstlane((int)g01),
                 (unsigned)__builtin_amdgcn_readfirstlane((int)g02),
                 (unsigned)__builtin_amdgcn_readfirstlane((int)g03) };
    i32x8 g1 = { __builtin_amdgcn_readfirstlane((int)g10),
                 __builtin_amdgcn_readfirstlane((int)g11),
                 __builtin_amdgcn_readfirstlane((int)g12),
                 __builtin_amdgcn_readfirstlane((int)g13),
                 __builtin_amdgcn_readfirstlane((int)g14),
                 __builtin_amdgcn_readfirstlane((int)g15),
                 __builtin_amdgcn_readfirstlane((int)g16),
                 __builtin_amdgcn_readfirstlane((int)g17) };
    i32x4 z4 = {0, 0, 0, 0};
#if __has_include(<hip/amd_detail/amd_gfx1250_TDM.h>)
    i32x8 z8 = {0, 0, 0, 0, 0, 0, 0, 0};
    __builtin_amdgcn_tensor_load_to_lds(g0, g1, z4, z4, z8, 0);
#else
    __builtin_amdgcn_tensor_load_to_lds(g0, g1, z4, z4, 0);
#endif
}
#endif

// ---------------- GEMM1: h = x @ W1 via v_wmma_f32_16x16x4_f32 ----------------
// One wave per 16-row tile; A tile double-buffered in LDS via TDM.
// Lanes 0-15 carry K={k,k+1}, lanes 16-31 K={k+2,k+3} for both A and B.

__global__ void k_gemm1(const float* __restrict__ x, const float* __restrict__ W,
                        float* __restrict__ h, int n) {
    int wslot = threadIdx.x >> 5;
    int wid   = blockIdx.x * G1_WAVES + wslot;
    int lane  = threadIdx.x & 31;
    int row0  = wid * 16;
    if (row0 >= n) return;                 // wave-uniform: EXEC stays all-1s for WMMA
    int lr = lane & 15;
    int hs = lane >> 4;
    int rows = n - row0; if (rows > 16) rows = 16;

    v8f acc = {0.f, 0.f, 0.f, 0.f, 0.f, 0.f, 0.f, 0.f};

#if USE_TDM
    __shared__ __align__(16) float lds_x[G1_WAVES][2][16 * RSTRIDE];
    const float* gtile = x + (size_t)row0 * FIN;
    tdm_load_tile16(gtile, (unsigned)(uintptr_t)&lds_x[wslot][0][0], n, rows);
    int cur = 0;
    for (int c = 0; c < NCHUNK; ++c) {
        if (c + 1 < NCHUNK) {
            // previous reads of the other buffer are done before we overwrite it
            asm volatile("s_wait_dscnt 0x0" ::: "memory");
            tdm_load_tile16(gtile + (size_t)(c + 1) * KCHUNK,
                            (unsigned)(uintptr_t)&lds_x[wslot][cur ^ 1][0], n, rows);
            __builtin_amdgcn_s_wait_tensorcnt((short)1);   // chunk c landed, c+1 in flight
        } else {
            __builtin_amdgcn_s_wait_tensorcnt((short)0);
        }
        const float* lx = &lds_x[wslot][cur][0] + (size_t)lr * RSTRIDE + 2 * hs;
        const float* wk = W + (size_t)(c * KCHUNK + 2 * hs) * HID + lr;
#pragma unroll
        for (int k = 0; k < KCHUNK; k += 4) {
            v2f a = *(const v2f*)(lx + k);          // ds_load_b64, bank-conflict-free
            v2f b;
            b.x = wk[(size_t)k * HID];              // W1[c*64+k+2hs][lr]
            b.y = wk[(size_t)k * HID + HID];
            acc = __builtin_amdgcn_wmma_f32_16x16x4_f32(
                      false, a, false, b, (short)0, acc, false, false);
        }
        cur ^= 1;
    }
#else
    // fallback: direct global gather (same math)
    int r = row0 + lr; if (r > n - 1) r = n - 1;
    const float* xrow = x + (size_t)r * FIN + 2 * hs;
    const float* wcol = W + (size_t)(2 * hs) * HID + lr;
#pragma unroll 8
    for (int k = 0; k < FIN; k += 4) {
        v2f a = *(const v2f*)(xrow + k);
        v2f b;
        b.x = wcol[(size_t)k * HID];
        b.y = wcol[(size_t)k * HID + HID];
        acc = __builtin_amdgcn_wmma_f32_16x16x4_f32(
                  false, a, false, b, (short)0, acc, false, false);
    }
#endif

    // D layout: lane lr holds col lr of rows hs*8 .. hs*8+7
    float* hp = h + (size_t)(row0 + hs * 8) * HID + lr;
    if (rows == 16) {
#pragma unroll
        for (int v = 0; v < 8; ++v) hp[(size_t)v * HID] = acc[v];
    } else {
#pragma unroll
        for (int v = 0; v < 8; ++v)
            if (hs * 8 + v < rows) hp[(size_t)v * HID] = acc[v];
    }
}

// ---------------- layer-1 aggregation ----------------

__global__ void k_self1(const float* __restrict__ h, const float* __restrict__ dinv,
                        float* __restrict__ agg, int n) {
    int idx = blockIdx.x * blockDim.x + threadIdx.x;
    if (idx < n * HID) {
        int i = idx >> 4;
        float di = dinv[i];
        agg[idx] = di * di * h[idx];       // self-loop: norm = dinv^2 * 1
    }
}

__global__ void k_edge1(const long long* __restrict__ ei, const float* __restrict__ w,
                        const float* __restrict__ dinv, const float* __restrict__ h,
                        float* __restrict__ agg, int E) {
    int e = blockIdx.x * blockDim.x + threadIdx.x;
    if (e >= E) return;
    int s = (int)ei[e];
    int d = (int)ei[(size_t)E + e];
    float norm = dinv[s] * w[e] * dinv[d];
    const float* hs_ = h + (size_t)s * HID;
    float* ad = agg + (size_t)d * HID;
    v4f h0 = *(const v4f*)(hs_ + 0);
    v4f h1 = *(const v4f*)(hs_ + 4);
    v4f h2 = *(const v4f*)(hs_ + 8);
    v4f h3 = *(const v4f*)(hs_ + 12);
#pragma unroll
    for (int c = 0; c < 4; ++c) atomicAdd(&ad[c],      norm * h0[c]);
#pragma unroll
    for (int c = 0; c < 4; ++c) atomicAdd(&ad[4 + c],  norm * h1[c]);
#pragma unroll
    for (int c = 0; c < 4; ++c) atomicAdd(&ad[8 + c],  norm * h2[c]);
#pragma unroll
    for (int c = 0; c < 4; ++c) atomicAdd(&ad[12 + c], norm * h3[c]);
}

// ---------------- GEMM2: t = relu(agg1 + b1) @ W2 (N padded 7->16) ----------------

__global__ void k_gemm2(const float* __restrict__ agg1, const float* __restrict__ b1,
                        const float* __restrict__ W2, float* __restrict__ t, int n) {
    int wid  = (blockIdx.x * blockDim.x + threadIdx.x) >> 5;
    int lane = threadIdx.x & 31;
    int row0 = wid * 16;
    if (row0 >= n) return;                 // wave-uniform
    int lr = lane & 15;
    int hs = lane >> 4;
    int rows = n - row0; if (rows > 16) rows = 16;
    int r  = row0 + lr; if (r > n - 1) r = n - 1;
    const float* arow = agg1 + (size_t)r * HID;

    v8f acc = {0.f, 0.f, 0.f, 0.f, 0.f, 0.f, 0.f, 0.f};
#pragma unroll
    for (int k = 0; k < HID; k += 4) {
        int c0 = k + 2 * hs;
        float a0 = arow[c0]     + b1[c0];
        float a1 = arow[c0 + 1] + b1[c0 + 1];
        v2f a;
        a.x = a0 > 0.f ? a0 : 0.f;         // fused bias + ReLU on A operand
        a.y = a1 > 0.f ? a1 : 0.f;
        v2f b;
        b.x = (lr < NCLS) ? W2[(size_t)c0 * NCLS + lr]       : 0.f;
        b.y = (lr < NCLS) ? W2[(size_t)(c0 + 1) * NCLS + lr] : 0.f;
        acc = __builtin_amdgcn_wmma_f32_16x16x4_f32(
                  false, a, false, b, (short)0, acc, false, false);
    }
    if (lr < TPAD) {
        float* tp = t + (size_t)(row0 + hs * 8) * TPAD + lr;
        if (rows == 16) {
#pragma unroll
            for (int v = 0; v < 8; ++v) tp[(size_t)v * TPAD] = (lr < NCLS) ? acc[v] : 0.f;
        } else {
#pragma unroll
            for (int v = 0; v < 8; ++v)
                if (hs * 8 + v < rows) tp[(size_t)v * TPAD] = (lr < NCLS) ? acc[v] : 0.f;
        }
    }
}

// ---------------- layer-2 aggregation ----------------

__global__ void k_self2(const float* __restrict__ t, const float* __restrict__ dinv,
                        float* __restrict__ agg, int n) {
    int idx = blockIdx.x * blockDim.x + threadIdx.x;
    if (idx < n * TPAD) {
        int i = idx >> 3;
        float di = dinv[i];
        agg[idx] = di * di * t[idx];
    }
}

__global__ void k_edge2(const long long* __restrict__ ei, const float* __restrict__ w,
                        const float* __restrict__ dinv, const float* __restrict__ t,
                        float* __restrict__ agg, int E) {
    int e = blockIdx.x * blockDim.x + threadIdx.x;
    if (e >= E) return;
    int s = (int)ei[e];
    int d = (int)ei[(size_t)E + e];
    float norm = dinv[s] * w[e] * dinv[d];
    const float* ts = t + (size_t)s * TPAD;
    float* ad = agg + (size_t)d * TPAD;
    v4f t0 = *(const v4f*)(ts + 0);
    v4f t1 = *(const v4f*)(ts + 4);
#pragma unroll
    for (int c = 0; c < 4; ++c) atomicAdd(&ad[c],     norm * t0[c]);
#pragma unroll
    for (int c = 0; c < 3; ++c) atomicAdd(&ad[4 + c], norm * t1[c]);  // col 7 is pad
}

// ---------------- bias + log_softmax ----------------

__global__ void k_final(const float* __restrict__ agg2, const float* __restrict__ b2,
                        float* __restrict__ out, int n) {
    int i = blockIdx.x * blockDim.x + threadIdx.x;
    if (i >= n) return;
    float v[NCLS];
    float m = -3.402823466e+38f;
#pragma unroll
    for (int c = 0; c < NCLS; ++c) {
        v[c] = agg2[(size_t)i * TPAD + c] + b2[c];
        m = fmaxf(m, v[c]);
    }
    float s = 0.f;
#pragma unroll
    for (int c = 0; c < NCLS; ++c) s += expf(v[c] - m);
    float lse = m + logf(s);
#pragma unroll
    for (int c = 0; c < NCLS; ++c) out[(size_t)i * NCLS + c] = v[c] - lse;
}

// ---------------- launch ----------------

extern "C" void kernel_launch(void* const* d_in, const int* in_sizes, int n_in,
                              void* d_out, int out_size, void* d_ws, size_t ws_size,
                              hipStream_t stream) {
    const float*     x  = (const float*)d_in[0];
    const long long* ei = (const long long*)d_in[1];
    const float*     w  = (const float*)d_in[2];
    const float*     W1 = (const float*)d_in[3];
    const float*     b1 = (const float*)d_in[4];
    const float*     W2 = (const float*)d_in[5];
    const float*     b2 = (const float*)d_in[6];
    float* out = (float*)d_out;

    int n = in_sizes[0] / FIN;
    int E = in_sizes[2];

    char*  ws  = (char*)d_ws;
    size_t off = 0;
    auto alloc = [&](size_t bytes) {
        void* p = ws + off;
        off = (off + bytes + 255) & ~(size_t)255;
        return p;
    };
    float* dinv = (float*)alloc((size_t)n * sizeof(float));
    float* h    = (float*)alloc((size_t)n * HID  * sizeof(float));
    float* agg1 = (float*)alloc((size_t)n * HID  * sizeof(float));
    float* t    = (float*)alloc((size_t)n * TPAD * sizeof(float));
    float* agg2 = (float*)alloc((size_t)n * TPAD * sizeof(float));

    const int B = 256;
    int ntiles = (n + 15) / 16;
    int g1_blocks = (ntiles + G1_WAVES - 1) / G1_WAVES;
    int g2_blocks = (ntiles + (B / 32) - 1) / (B / 32);

    k_init_deg <<<(n + B - 1) / B, B, 0, stream>>>(dinv, n);
    k_deg_edges<<<(E + B - 1) / B, B, 0, stream>>>(ei, w, dinv, E);
    k_dinv     <<<(n + B - 1) / B, B, 0, stream>>>(dinv, n);

    k_gemm1    <<<g1_blocks, G1_BLOCK, 0, stream>>>(x, W1, h, n);
    k_self1    <<<((size_t)n * HID + B - 1) / B, B, 0, stream>>>(h, dinv, agg1, n);
    k_edge1    <<<(E + B - 1) / B, B, 0, stream>>>(ei, w, dinv, h, agg1, E);

    k_gemm2    <<<g2_blocks, B, 0, stream>>>(agg1, b1, W2, t, n);
    k_self2    <<<((size_t)n * TPAD + B - 1) / B, B, 0, stream>>>(t, dinv, agg2, n);
    k_edge2    <<<(E + B - 1) / B, B, 0, stream>>>(ei, w, dinv, t, agg2, E);

    k_final    <<<(n + B - 1) / B, B, 0, stream>>>(agg2, b2, out, n);
}